// EquivariantConv_65309272703462
// MI455X (gfx1250) — compile-verified
//
#include <hip/hip_runtime.h>
#include <hip/hip_bf16.h>

typedef float v2f __attribute__((ext_vector_type(2)));
typedef float v8f __attribute__((ext_vector_type(8)));

#define BB  4
#define NN  4096
#define VV  4096
#define PP  16
#define CC  16
#define NKK 4

#if __has_builtin(__builtin_amdgcn_wmma_f32_16x16x4_f32)
#define HAVE_WMMA_F32 1
#else
#define HAVE_WMMA_F32 0
#endif

// ---------------- compile-time Clebsch-Gordan tables (Racah formula) --------
constexpr double cfact(int n) { double r = 1.0; for (int i = 2; i <= n; ++i) r *= i; return r; }
constexpr double csqrt_(double x) {
  if (x <= 0.0) return 0.0;
  double g = x > 1.0 ? x : 1.0;
  for (int i = 0; i < 100; ++i) { double ng = 0.5 * (g + x / g); if (ng == g) break; g = ng; }
  return g;
}
constexpr int imax3(int a, int b, int c) { int m = a; if (b > m) m = b; if (c > m) m = c; return m; }
constexpr int imin3(int a, int b, int c) { int m = a; if (b < m) m = b; if (c < m) m = c; return m; }

constexpr double cgcoef(int j1, int m1, int j2, int m2, int J, int M) {
  if (M != m1 + m2) return 0.0;
  double pref = (2.0 * J + 1.0) * cfact(j1 + j2 - J) * cfact(j1 - j2 + J) *
                cfact(-j1 + j2 + J) / cfact(j1 + j2 + J + 1);
  pref *= cfact(J + M) * cfact(J - M) * cfact(j1 - m1) * cfact(j1 + m1) *
          cfact(j2 - m2) * cfact(j2 + m2);
  const int tlo = imax3(0, j2 - J - m1, j1 + m2 - J);
  const int thi = imin3(j1 + j2 - J, j1 - m1, j2 + m2);
  double s = 0.0;
  for (int t = tlo; t <= thi; ++t) {
    double d = cfact(t) * cfact(j1 + j2 - J - t) * cfact(j1 - m1 - t) *
               cfact(j2 + m2 - t) * cfact(J - j2 + m1 + t) * cfact(J - j1 - m2 + t);
    s += ((t & 1) ? -1.0 : 1.0) / d;
  }
  return csqrt_(pref) * s;
}

// 15 output blocks, in reference concat order:
// y0: (l0,k0) (1,1,J0) (2,2,J0)
// y1: (l0,k1) (1,0) (1,1) (1,2) (2,1) (2,2)   all J=1
// y2: (l0,k2) (1,1) (1,2) (2,0) (2,1) (2,2)   all J=2
// (l=0 "direct" blocks equal the CG path since <0 0; k n | k M> = delta(n,M).)
constexpr int H_BL_J[15] = {0,0,0, 1,1,1,1,1,1, 2,2,2,2,2,2};
constexpr int H_BL_L[15] = {0,1,2, 0,1,1,1,2,2, 0,1,1,2,2,2};
constexpr int H_BL_K[15] = {0,1,2, 1,0,1,2,1,2, 2,1,2,0,1,2};

struct QData { float q[615]; };

constexpr QData buildQ() {
  QData d{};
  int off = 0;
  for (int b = 0; b < 15; ++b) {
    const int J = H_BL_J[b], l = H_BL_L[b], k = H_BL_K[b];
    for (int Jm = 0; Jm < 2 * J + 1; ++Jm)
      for (int mi = 0; mi < 2 * l + 1; ++mi)
        for (int ki = 0; ki < 2 * k + 1; ++ki) {
          const int M = Jm - J, m = mi - l, n = ki - k;
          d.q[off + (Jm * (2 * l + 1) + mi) * (2 * k + 1) + ki] =
              (M == m + n) ? (float)cgcoef(l, m, k, n, J, M) : 0.0f;
        }
    off += (2 * J + 1) * (2 * l + 1) * (2 * k + 1);
  }
  return d;
}

__constant__ QData QTAB = buildQ();
__constant__ int BL_J[15]    = {0,0,0, 1,1,1,1,1,1, 2,2,2,2,2,2};
__constant__ int BL_L[15]    = {0,1,2, 0,1,1,1,2,2, 0,1,1,2,2,2};
__constant__ int BL_K[15]    = {0,1,2, 1,0,1,2,1,2, 2,1,2,0,1,2};
__constant__ int BL_POS[15]  = {0,1,2, 0,1,2,3,4,5, 0,1,2,3,4,5};
__constant__ int BL_QOFF[15] = {0,1,10, 35,44,53,80,125,170, 245,270,315,390,415,490};
__constant__ int JOBSTART[16] = {0,64,128, 192,384,576,768,960,1152,
                                 1344,1664,1984,2304,2624,2944, 3264};

#define YW 52  // padded Y row stride (breaks the hi-half bank aliasing of 48)

__global__ __launch_bounds__(128)
void equi_conv_kernel(const float* __restrict__ sig0, const float* __restrict__ ker0,
                      const float* __restrict__ sig1, const float* __restrict__ ker1,
                      const float* __restrict__ sig2, const float* __restrict__ ker2,
                      const int* __restrict__ pidx, float* __restrict__ out) {
  __shared__ float sS[16][144];   // [p][row], row=(l,m,c): l0:0-15, l1:16-63, l2:64-143
  __shared__ float sK[16][48];    // [p][col], col: k0:0-3, k1:4-15, k2:16-35, pad:36-47
  __shared__ float sY[144][YW];   // Y = S^T * K  (144 x 36 valid, padded)
  __shared__ int   sPb[16], sPn[16];

  const int site = blockIdx.x;  // site = b*V + v
  const int tid  = threadIdx.x;

  if (tid < PP) {
    sPb[tid] = pidx[(site * PP + tid) * 2 + 0];
    sPn[tid] = pidx[(site * PP + tid) * 2 + 1];
  }
  __syncthreads();

  // ---- Phase 0: cooperative gather of S (signals at patch indices) and K ----
  for (int i = tid; i < PP * 144; i += 128) {
    const int p = i / 144, row = i % 144;
    const size_t bn = (size_t)sPb[p] * NN + sPn[p];
    float v;
    if (row < 16) {
      v = sig0[bn * CC + row];
    } else if (row < 64) {
      const int m = (row - 16) >> 4, c = row & 15;
      v = sig1[(bn * 3 + m) * CC + c];
    } else {
      const int m = (row - 64) >> 4, c = row & 15;
      v = sig2[(bn * 5 + m) * CC + c];
    }
    sS[p][row] = v;
  }
  for (int i = tid; i < PP * 48; i += 128) {
    const int p = i / 48, col = i % 48;
    const size_t base = (size_t)site * PP + p;
    float v;
    if (col < 4)       v = ker0[base * (1 * NKK) + col];
    else if (col < 16) v = ker1[base * (3 * NKK) + (col - 4)];
    else if (col < 36) v = ker2[base * (5 * NKK) + (col - 16)];
    else               v = 0.0f;  // pad so WMMA garbage columns are exactly 0
    sK[p][col] = v;
  }
  __syncthreads();

  // ---- Phase 1: Y[144x36] = S^T(144x16) * K(16x36) via f32 WMMA 16x16x4 ----
  // 9 M-tiles x 3 N-tiles = 27 tile jobs split over the block's 4 waves.
  const int wv = tid >> 5, lane = tid & 31, hi = lane >> 4, li = lane & 15;
  for (int job = wv; job < 27; job += 4) {
    const int mt = job / 3, nt = job % 3;
    v8f acc = {0.f, 0.f, 0.f, 0.f, 0.f, 0.f, 0.f, 0.f};
#if HAVE_WMMA_F32
    // 32-bit A 16x4 layout: lanes0-15 VGPR0/1 = K0/K1, lanes16-31 = K2/K3 (M=lane%16).
    // B assumed symmetric (N=lane%16). K=16 -> chain 4 WMMAs.
#pragma unroll
    for (int j = 0; j < 4; ++j) {
      const int k0 = 4 * j + 2 * hi;
      v2f a, bm;
      a.x  = sS[k0][mt * 16 + li];
      a.y  = sS[k0 + 1][mt * 16 + li];
      bm.x = sK[k0][nt * 16 + li];
      bm.y = sK[k0 + 1][nt * 16 + li];
      acc = __builtin_amdgcn_wmma_f32_16x16x4_f32(false, a, false, bm,
                                                  (short)0, acc, false, false);
    }
#else
    // scalar fallback producing the identical D-fragment layout
    for (int kp = 0; kp < 16; ++kp) {
      const float bv = sK[kp][nt * 16 + li];
#pragma unroll
      for (int r = 0; r < 8; ++r)
        acc[r] += sS[kp][mt * 16 + r + 8 * hi] * bv;
    }
#endif
    // D layout: VGPR r -> row r (lanes 0-15) / row r+8 (lanes 16-31), col = lane%16
#pragma unroll
    for (int r = 0; r < 8; ++r)
      sY[mt * 16 + r + 8 * hi][nt * 16 + li] = acc[r];
  }
  __syncthreads();

  // ---- Phase 2: CG contraction Y -> 15 output blocks (3264 floats/site) ----
  for (int i = tid; i < 3264; i += 128) {
    int blk = 0;
    while (i >= JOBSTART[blk + 1]) ++blk;
    const int loc = i - JOBSTART[blk];
    const int J = BL_J[blk], l = BL_L[blk], k = BL_K[blk];
    const int Jm = loc >> 6, ch = loc & 63, n = ch >> 4, c = ch & 15;
    const int dl = 2 * l + 1, dk = 2 * k + 1;
    const int rb = (l == 0) ? 0 : (l == 1) ? 16 : 64;
    const int cb = (k == 0) ? 0 : (k == 1) ? 4 : 16;
    const float* q = &QTAB.q[BL_QOFF[blk] + Jm * dl * dk];
    float acc = 0.f;
    for (int m = 0; m < dl; ++m)
      for (int kk = 0; kk < dk; ++kk)
        acc += q[m * dk + kk] * sY[rb + m * 16 + c][cb + kk * NKK + n];
    const size_t width  = (J == 0) ? 192 : 384;
    const size_t outoff = (J == 0) ? 0
                        : (J == 1) ? (size_t)BB * VV * 192
                                   : (size_t)BB * VV * 192 + (size_t)BB * VV * 3 * 384;
    out[outoff + ((size_t)site * (2 * J + 1) + Jm) * width +
        (size_t)BL_POS[blk] * 64 + ch] = acc;
  }
}

extern "C" void kernel_launch(void* const* d_in, const int* in_sizes, int n_in,
                              void* d_out, int out_size, void* d_ws, size_t ws_size,
                              hipStream_t stream) {
  (void)in_sizes; (void)n_in; (void)d_ws; (void)ws_size; (void)out_size;
  // setup_inputs order: signal_0, kernel_0, signal_1, kernel_1, signal_2, kernel_2, patches_idx
  const float* sig0 = (const float*)d_in[0];
  const float* ker0 = (const float*)d_in[1];
  const float* sig1 = (const float*)d_in[2];
  const float* ker1 = (const float*)d_in[3];
  const float* sig2 = (const float*)d_in[4];
  const float* ker2 = (const float*)d_in[5];
  const int*   pidx = (const int*)d_in[6];
  float* out = (float*)d_out;

  dim3 grid(BB * VV), block(128);
  hipLaunchKernelGGL(equi_conv_kernel, grid, block, 0, stream,
                     sig0, ker0, sig1, ker1, sig2, ker2, pidx, out);
}